// MetaEmbedding_Classifier_89532888252518
// MI455X (gfx1250) — compile-verified
//
#include <hip/hip_runtime.h>
#include <hip/hip_bf16.h>
#include <math.h>

typedef __attribute__((ext_vector_type(2))) float v2f;
typedef __attribute__((ext_vector_type(8))) float v8f;
typedef int b128_t __attribute__((vector_size(16)));   // payload type of async b128 builtin

#define TM 64
#define TN 64
#define TK 64
#define LDSS 68   // padded LDS row stride (floats): 68 mod 64 = 4 -> spreads banks

// ---- CDNA5 async global->LDS staging (ASYNCcnt path), fallback to float4 ----
#if defined(__has_builtin)
#  if __has_builtin(__builtin_amdgcn_global_load_async_to_lds_b128)
#    define USE_ASYNC_LDS 1
#  endif
#endif
#ifndef USE_ASYNC_LDS
#  define USE_ASYNC_LDS 0
#endif

__device__ __forceinline__ void stage16(const float* __restrict__ g, float* l)
{
#if USE_ASYNC_LDS
    __builtin_amdgcn_global_load_async_to_lds_b128(
        (__attribute__((address_space(1))) b128_t*)g,
        (__attribute__((address_space(3))) b128_t*)l,
        0, 0);
#else
    *(float4*)l = *(const float4*)g;
#endif
}

__device__ __forceinline__ void stage_fence()
{
#if USE_ASYNC_LDS
    asm volatile("s_wait_asynccnt 0x0" ::: "memory");
#endif
    __syncthreads();
}

// ---------------------------------------------------------------------------
// NT GEMM: C[i,n] = (sum_k A[i,k] * W[n,k] + bias[n]) * (scale*colscale[n])
// A may be split into (A0|A1) along K at Ksplit (virtual concat).
// Block: 256 thr = 8 waves; block tile 64x64; per-wave 16x32 (2 accumulators).
// OOB B-rows are clamped (not zeroed): they only feed discarded C columns.
// ---------------------------------------------------------------------------
__global__ __launch_bounds__(256) void gemm_nt_wmma(
    const float* __restrict__ A0, const float* __restrict__ A1, int Ksplit,
    const float* __restrict__ W, int ldw,
    float* __restrict__ C, int ldc,
    int M, int N, int K,
    const float* __restrict__ bias, const float* __restrict__ colscale,
    float scale)
{
    __shared__ float sA[TM * LDSS];
    __shared__ float sB[TN * LDSS];

    const int tid  = threadIdx.x;
    const int lane = tid & 31;
    const int wave = tid >> 5;
    const int wm   = wave >> 1;      // 0..3  (16-row slab)
    const int wn   = wave & 1;       // 0..1  (32-col slab)
    const int m0   = blockIdx.x * TM;
    const int n0   = blockIdx.y * TN;
    const int fr   = lane & 15;      // fragment row (A: M, B/C: N)
    const int half = lane >> 4;      // K-half select / C row offset

    v8f acc0 = {};
    v8f acc1 = {};

    for (int k0 = 0; k0 < K; k0 += TK) {
        const float* Abase = (k0 < Ksplit) ? A0 : A1;
        const int    kbase = (k0 < Ksplit) ? k0 : (k0 - Ksplit);

        // stage A tile 64x64 and B tile 64x64 (4 x b128 per thread each)
        #pragma unroll
        for (int it = 0; it < 4; ++it) {
            int p  = tid + it * 256;       // 0..1023 float4 slots
            int r  = p >> 4;               // 0..63
            int c4 = (p & 15) << 2;        // 0..60
            stage16(Abase + (size_t)(m0 + r) * Ksplit + kbase + c4,
                    &sA[r * LDSS + c4]);
            int nrow = n0 + r; if (nrow >= N) nrow = N - 1;   // clamp, see note
            stage16(W + (size_t)nrow * ldw + k0 + c4,
                    &sB[r * LDSS + c4]);
        }
        // prefetch next K chunk (global_prefetch_b8)
        {
            int kn = k0 + TK;
            if (kn < K) {
                const float* An = (kn < Ksplit) ? A0 : A1;
                int kb = (kn < Ksplit) ? kn : (kn - Ksplit);
                __builtin_prefetch(An + (size_t)(m0 + (tid >> 4)) * Ksplit + kb + ((tid & 15) << 2), 0, 0);
                int nr = n0 + (tid >> 4); if (nr >= N) nr = N - 1;
                __builtin_prefetch(W + (size_t)nr * ldw + kn + ((tid & 15) << 2), 0, 0);
            }
        }
        stage_fence();

        const float* pa  = &sA[(wm * 16      + fr) * LDSS + 2 * half];
        const float* pb0 = &sB[(wn * 32      + fr) * LDSS + 2 * half];
        const float* pb1 = &sB[(wn * 32 + 16 + fr) * LDSS + 2 * half];
        #pragma unroll
        for (int kk = 0; kk < TK; kk += 4) {
            v2f a  = *(const v2f*)(pa  + kk);
            v2f b0 = *(const v2f*)(pb0 + kk);
            v2f b1 = *(const v2f*)(pb1 + kk);
            acc0 = __builtin_amdgcn_wmma_f32_16x16x4_f32(
                false, a, false, b0, (short)0, acc0, false, false);
            acc1 = __builtin_amdgcn_wmma_f32_16x16x4_f32(
                false, a, false, b1, (short)0, acc1, false, false);
        }
        __syncthreads();
    }

    #pragma unroll
    for (int t = 0; t < 2; ++t) {
        const v8f* accp = t ? &acc1 : &acc0;
        int ncol = n0 + wn * 32 + t * 16 + fr;
        int nclamp = (ncol < N) ? ncol : 0;
        float badd = bias ? bias[nclamp] : 0.f;
        float cmul = colscale ? (scale * colscale[nclamp]) : 1.f;
        #pragma unroll
        for (int v = 0; v < 8; ++v) {
            int grow = m0 + wm * 16 + v + 8 * half;
            float val = ((*accp)[v] + badd) * cmul;
            if (ncol < N)
                C[(size_t)grow * ldc + ncol] = val;
        }
    }
}

// ---------------------------------------------------------------------------
// NN GEMM (memory_feature) with fused blend epilogue:
//   mf[i,n]   = sum_k att[i,k] * cent[k,n]      (k >= Krows -> cent treated 0)
//   Fout[i,n] = (1-p[i]) * feat[i,n] + p[i] * mf[i,n]
// B staged transposed through the synchronous path (needs true zero padding
// on the reduction dim); A staged asynchronously.
// ---------------------------------------------------------------------------
__global__ __launch_bounds__(256) void gemm_nn_fuse_wmma(
    const float* __restrict__ Aatt,   // M x K (K padded w/ zeros)
    const float* __restrict__ Cent,   // Krows x N
    const float* __restrict__ feat,   // M x N
    const float* __restrict__ pvec,   // M
    float* __restrict__ Fout,         // M x N
    int M, int N, int K, int Krows)
{
    __shared__ float sA [TM * LDSS];
    __shared__ float sBT[TN * LDSS];  // transposed: sBT[n][k]

    const int tid  = threadIdx.x;
    const int lane = tid & 31;
    const int wave = tid >> 5;
    const int wm   = wave >> 1;
    const int wn   = wave & 1;
    const int m0   = blockIdx.x * TM;
    const int n0   = blockIdx.y * TN;
    const int fr   = lane & 15;
    const int half = lane >> 4;

    v8f acc0 = {};
    v8f acc1 = {};

    for (int k0 = 0; k0 < K; k0 += TK) {
        #pragma unroll
        for (int it = 0; it < 4; ++it) {
            int p  = tid + it * 256;
            int r  = p >> 4;
            int c4 = (p & 15) << 2;
            stage16(Aatt + (size_t)(m0 + r) * K + k0 + c4,
                    &sA[r * LDSS + c4]);
        }
        // stage B transposed: sBT[n][k] = Cent[k0+k][n0+n]; zero beyond Krows
        {
            int n = tid & 63;            // 0..63
            int kb = tid >> 6;           // 0..3
            #pragma unroll
            for (int it = 0; it < 16; ++it) {
                int kr = kb + it * 4;    // 0..63
                int kg = k0 + kr;
                float v = (kg < Krows) ? Cent[(size_t)kg * N + n0 + n] : 0.f;
                sBT[n * LDSS + kr] = v;
            }
        }
        stage_fence();

        const float* pa  = &sA [(wm * 16      + fr) * LDSS + 2 * half];
        const float* pb0 = &sBT[(wn * 32      + fr) * LDSS + 2 * half];
        const float* pb1 = &sBT[(wn * 32 + 16 + fr) * LDSS + 2 * half];
        #pragma unroll
        for (int kk = 0; kk < TK; kk += 4) {
            v2f a  = *(const v2f*)(pa  + kk);
            v2f b0 = *(const v2f*)(pb0 + kk);
            v2f b1 = *(const v2f*)(pb1 + kk);
            acc0 = __builtin_amdgcn_wmma_f32_16x16x4_f32(
                false, a, false, b0, (short)0, acc0, false, false);
            acc1 = __builtin_amdgcn_wmma_f32_16x16x4_f32(
                false, a, false, b1, (short)0, acc1, false, false);
        }
        __syncthreads();
    }

    #pragma unroll
    for (int t = 0; t < 2; ++t) {
        const v8f* accp = t ? &acc1 : &acc0;
        int ncol = n0 + wn * 32 + t * 16 + fr;
        #pragma unroll
        for (int v = 0; v < 8; ++v) {
            int grow = m0 + wm * 16 + v + 8 * half;
            float p = pvec[grow];
            float f = feat[(size_t)grow * N + ncol];
            Fout[(size_t)grow * N + ncol] = (1.f - p) * f + p * (*accp)[v];
        }
    }
}

// ---------------------------------------------------------------------------
// Per-relation norms: cc[j] = ||centroids_j||^2 ; wscale[j] = 1/||cls_w_j||
// ---------------------------------------------------------------------------
__global__ __launch_bounds__(64) void row_norms_kernel(
    const float* __restrict__ cent, const float* __restrict__ clsw,
    float* __restrict__ cc, float* __restrict__ wscale, int D)
{
    int j = blockIdx.x;
    int lane = threadIdx.x & 31;
    int wave = threadIdx.x >> 5;
    const float* src = (wave == 0) ? cent : clsw;
    float s = 0.f;
    for (int k = lane; k < D; k += 32) {
        float x = src[(size_t)j * D + k];
        s += x * x;
    }
    #pragma unroll
    for (int off = 16; off > 0; off >>= 1)
        s += __shfl_xor(s, off, 32);
    if (lane == 0) {
        if (wave == 0) cc[j] = s;
        else           wscale[j] = 1.f / sqrtf(s);
    }
}

// ---------------------------------------------------------------------------
// Per-sample middle stage: ||x||^2, min/argmin dist, softmax->att (zeroed at
// relation), sigmoid gate p. One 256-thread block per sample.
// ---------------------------------------------------------------------------
__global__ __launch_bounds__(256) void rowwise_mid_kernel(
    const float* __restrict__ x,      // B x D (feat)
    const float* __restrict__ Gatt,   // B x 512 (x @ all_rel^T), cols<500 valid
    const float* __restrict__ Gcent,  // B x 512 (x @ cent^T),    cols<500 valid
    const float* __restrict__ cc,     // 500
    const int*   __restrict__ relation,
    const float* __restrict__ selw,   // D
    const float* __restrict__ selb,   // 1
    float* __restrict__ att,          // B x 512 out (may alias Gatt)
    float* __restrict__ pvec,         // B
    float* __restrict__ outv,         // values_nn0
    float* __restrict__ outl,         // labels_nn0 (as float)
    int D, int R)
{
    const int i = blockIdx.x;
    const int tid = threadIdx.x;
    __shared__ float red [256];
    __shared__ float red2[256];
    __shared__ int   redi[256];

    // 1) ||x||^2 and x . sel_w
    float sx = 0.f, sd = 0.f;
    for (int k = tid; k < D; k += 256) {
        float xv = x[(size_t)i * D + k];
        sx += xv * xv;
        sd += xv * selw[k];
    }
    red[tid] = sx; red2[tid] = sd; __syncthreads();
    for (int s = 128; s > 0; s >>= 1) {
        if (tid < s) { red[tid] += red[tid + s]; red2[tid] += red2[tid + s]; }
        __syncthreads();
    }
    sx = red[0]; sd = red2[0];
    __syncthreads();

    // 2) min / argmin over dist(x, centroids[3:]) with first-index tie-break
    float bestv = 3.402823466e+38f;
    int   besti = 0x7fffffff;
    for (int j = 3 + tid; j < R; j += 256) {
        float d2 = sx - 2.f * Gcent[(size_t)i * 512 + j] + cc[j];
        float dist = sqrtf(fmaxf(d2, 1e-12f));
        if (dist < bestv || (dist == bestv && j < besti)) { bestv = dist; besti = j; }
    }
    red[tid] = bestv; redi[tid] = besti; __syncthreads();
    for (int s = 128; s > 0; s >>= 1) {
        if (tid < s) {
            float ov = red[tid + s]; int oi = redi[tid + s];
            if (ov < red[tid] || (ov == red[tid] && oi < redi[tid])) {
                red[tid] = ov; redi[tid] = oi;
            }
        }
        __syncthreads();
    }
    if (tid == 0) { outv[i] = red[0]; outl[i] = (float)redi[0]; }
    __syncthreads();

    // 3) softmax over R, then zero att[i, relation[i]]
    float mx = -3.402823466e+38f;
    for (int j = tid; j < R; j += 256)
        mx = fmaxf(mx, Gatt[(size_t)i * 512 + j]);
    red[tid] = mx; __syncthreads();
    for (int s = 128; s > 0; s >>= 1) {
        if (tid < s) red[tid] = fmaxf(red[tid], red[tid + s]);
        __syncthreads();
    }
    mx = red[0]; __syncthreads();

    float ssum = 0.f;
    for (int j = tid; j < R; j += 256)
        ssum += expf(Gatt[(size_t)i * 512 + j] - mx);
    red[tid] = ssum; __syncthreads();
    for (int s = 128; s > 0; s >>= 1) {
        if (tid < s) red[tid] += red[tid + s];
        __syncthreads();
    }
    float invS = 1.f / red[0];
    __syncthreads();

    int rel = relation[i];
    for (int j = tid; j < 512; j += 256) {
        float a = 0.f;
        if (j < R && j != rel)
            a = expf(Gatt[(size_t)i * 512 + j] - mx) * invS;
        att[(size_t)i * 512 + j] = a;
    }
    if (tid == 0)
        pvec[i] = 1.f / (1.f + expf(-(sd + selb[0])));
}

// ---------------------------------------------------------------------------
// ex = fused / (1 + ||fused||) per row
// ---------------------------------------------------------------------------
__global__ __launch_bounds__(256) void rowwise_ex_kernel(
    const float* __restrict__ F, float* __restrict__ EX, int D)
{
    const int i = blockIdx.x;
    const int tid = threadIdx.x;
    __shared__ float red[256];
    float s = 0.f;
    for (int k = tid; k < D; k += 256) {
        float f = F[(size_t)i * D + k];
        s += f * f;
    }
    red[tid] = s; __syncthreads();
    for (int st = 128; st > 0; st >>= 1) {
        if (tid < st) red[tid] += red[tid + st];
        __syncthreads();
    }
    float inv = 1.f / (1.f + sqrtf(red[0]));
    __syncthreads();
    for (int k = tid; k < D; k += 256)
        EX[(size_t)i * D + k] = F[(size_t)i * D + k] * inv;
}

// ---------------------------------------------------------------------------
extern "C" void kernel_launch(void* const* d_in, const int* in_sizes, int n_in,
                              void* d_out, int out_size, void* d_ws, size_t ws_size,
                              hipStream_t stream)
{
    (void)in_sizes; (void)n_in; (void)out_size; (void)ws_size;

    const int B = 1024, D = 512, R = 500;

    const float* head  = (const float*)d_in[0];
    const float* tail  = (const float*)d_in[1];
    const int*   rel   = (const int*)  d_in[2];
    const float* arel  = (const float*)d_in[3];
    const float* cent  = (const float*)d_in[4];
    const float* fc2w  = (const float*)d_in[5];
    const float* fc2b  = (const float*)d_in[6];
    const float* selw  = (const float*)d_in[7];
    const float* selb  = (const float*)d_in[8];
    const float* clsw  = (const float*)d_in[9];

    float* out    = (float*)d_out;
    float* logits = out;                         // B*R
    float* feat   = out + (size_t)B * R;         // B*D
    float* valv   = feat + (size_t)B * D;        // B
    float* labl   = valv + B;                    // B

    float* ws     = (float*)d_ws;
    float* W0     = ws;                          // B*512 : Gatt -> att -> EX
    float* W1     = W0 + (size_t)B * 512;        // B*512 : Gcent -> fused
    float* pvec   = W1 + (size_t)B * 512;        // B
    float* cc     = pvec + B;                    // 512
    float* wscale = cc + 512;                    // 512

    dim3 blk(256);
    dim3 gN512(B / TM, D / TN);                  // 16 x 8
    dim3 gN500(B / TM, (R + TN - 1) / TN);       // 16 x 8

    // Precompute per-relation norms.
    row_norms_kernel<<<R, 64, 0, stream>>>(cent, clsw, cc, wscale, D);

    // feat = concat(head,tail) @ fc2_w^T + fc2_b       (M=1024,N=512,K=1024)
    gemm_nt_wmma<<<gN512, blk, 0, stream>>>(head, tail, D, fc2w, 2 * D,
                                            feat, D, B, D, 2 * D,
                                            fc2b, nullptr, 1.f);

    // Gatt = feat @ all_relation^T                     (N=500,K=512)
    gemm_nt_wmma<<<gN500, blk, 0, stream>>>(feat, nullptr, D, arel, D,
                                            W0, 512, B, R, D,
                                            nullptr, nullptr, 1.f);

    // Gcent = feat @ centroids^T                       (N=500,K=512)
    gemm_nt_wmma<<<gN500, blk, 0, stream>>>(feat, nullptr, D, cent, D,
                                            W1, 512, B, R, D,
                                            nullptr, nullptr, 1.f);

    // per-sample: min/argmin dist, softmax -> att (in W0), gate p
    rowwise_mid_kernel<<<B, blk, 0, stream>>>(feat, W0, W1, cc, rel, selw, selb,
                                              W0, pvec, valv, labl, D, R);

    // fused = (1-p)*feat + p*(att @ centroids)  -> W1  (K padded to 512)
    gemm_nn_fuse_wmma<<<gN512, blk, 0, stream>>>(W0, cent, feat, pvec, W1,
                                                 B, D, 512, R);

    // EX = fused / (1 + ||fused||)              -> W0
    rowwise_ex_kernel<<<B, blk, 0, stream>>>(W1, W0, D);

    // logits = 16 * (EX @ (cls_w/||cls_w||)^T)         (N=500,K=512, ldc=500)
    gemm_nt_wmma<<<gN500, blk, 0, stream>>>(W0, nullptr, D, clsw, D,
                                            logits, R, B, R, D,
                                            nullptr, wscale, 16.f);
}